// SparseConv3d_21560735826242
// MI455X (gfx1250) — compile-verified
//
#include <hip/hip_runtime.h>

typedef float v2f __attribute__((ext_vector_type(2)));
typedef float v8f __attribute__((ext_vector_type(8)));

#define IN_C   8
#define OUT_C  8
#define DHW    160
#define TILE_X 16           // M dimension: 16 output pixels along x
#define TILE_Y 8            // 8 waves per block, one y-row each
#define HALO_Y (TILE_Y + 2)
#define LDS_XW 20           // 16 + 2 halo + 2 pad (px+3 reads stay in-row)
#define WROW   288          // 72 steps * 4 (kx padded 3 -> 4)

__global__ __launch_bounds__(256)
void conv3d_wmma_f32_kernel(const float* __restrict__ x,
                            const float* __restrict__ w,
                            const float* __restrict__ bias,
                            float* __restrict__ out)
{
    __shared__ float xlds[IN_C * 3 * HALO_Y * LDS_XW];   // 4800 f = 19.2 KB
    __shared__ float wlds[16 * WROW];                    // 4608 f = 18.4 KB
    __shared__ float slds[TILE_Y][OUT_C][TILE_X];        // 1024 f =  4.0 KB

    const int tid = threadIdx.x;
    const int xt  = blockIdx.x % (DHW / TILE_X);
    const int yt  = blockIdx.x / (DHW / TILE_X);
    const int z   = blockIdx.y;
    const int n   = blockIdx.z;
    const int x0  = xt * TILE_X;
    const int y0  = yt * TILE_Y;

    // ---- stage weights: wlds[oc16][step*4 + kx]; rows oc>=8 and slot kx==3
    //      are zero, which makes the padded K-slots of A don't-care. ----
    for (int idx = tid; idx < 16 * WROW; idx += 256) {
        int oc = idx / WROW;
        int r  = idx % WROW;
        int s  = r >> 2;
        int j  = r & 3;          // kx slot (3 == pad)
        float v = 0.0f;
        if (oc < OUT_C && j < 3) {
            int ic = s / 9;
            int kz = (s / 3) % 3;
            int ky = s % 3;
            v = w[(((oc * IN_C + ic) * 3 + kz) * 3 + ky) * 3 + j];
        }
        wlds[idx] = v;
    }

    // ---- stage haloed input tile into LDS (zero-fill out of bounds;
    //      every slot is written, so A reads are always finite). ----
    const long plane = (long)DHW * DHW;
    const float* xn  = x + (long)n * IN_C * DHW * plane;
    for (int idx = tid; idx < IN_C * 3 * HALO_Y * LDS_XW; idx += 256) {
        int xi = idx % LDS_XW;
        int dy = (idx / LDS_XW) % HALO_Y;
        int dz = (idx / (LDS_XW * HALO_Y)) % 3;
        int ic = idx / (LDS_XW * HALO_Y * 3);
        int gz = z  + dz - 1;
        int gy = y0 + dy - 1;
        int gx = x0 + xi - 1;
        float v = 0.0f;
        if (gz >= 0 && gz < DHW && gy >= 0 && gy < DHW && gx >= 0 && gx < DHW)
            v = xn[(long)ic * DHW * plane + (long)gz * plane + gy * DHW + gx];
        xlds[idx] = v;
    }
    __syncthreads();

    const int  lane   = tid & 31;
    const int  wv     = tid >> 5;       // wave id -> y row 0..7
    const int  px     = lane & 15;      // M index (low half)
    const bool hihalf = lane >= 16;     // lanes 16-31 hold K-slots {2,3}
    const int  hioff  = hihalf ? 2 : 0;
    const int  oc     = lane & 15;      // N index

    // bias-initialized accumulator (oc>=8 columns are dead)
    float bv = bias[oc & 7];
    bv = (oc < OUT_C) ? bv : 0.0f;
    v8f c;
    #pragma unroll
    for (int i = 0; i < 8; ++i) c[i] = bv;

    // ---- implicit GEMM: 72 x V_WMMA_F32_16X16X4_F32, K padded 216->288.
    //      No conditionals in the loop: padded K-slots are killed by the
    //      zeroed B rows, so A loads are unconditional (no EXEC edits). ----
    const int arow0 = wv * LDS_XW + px + hioff;
    const int wcol0 = oc * WROW + hioff;
    for (int ic = 0; ic < IN_C; ++ic) {
        #pragma unroll
        for (int kz = 0; kz < 3; ++kz) {
            const int rowbase = ((ic * 3 + kz) * HALO_Y) * LDS_XW + arow0;
            const int wbase   = wcol0 + ic * 36 + kz * 12;
            #pragma unroll
            for (int ky = 0; ky < 3; ++ky) {
                const int ao = rowbase + ky * LDS_XW;
                v2f a;
                a.x = xlds[ao];       // kx = 0 (lo lanes) / 2 (hi lanes)
                a.y = xlds[ao + 1];   // kx = 1 (lo)      / pad (hi, B row = 0)
                const int wo = wbase + ky * 4;
                v2f b;
                b.x = wlds[wo];
                b.y = wlds[wo + 1];
                c = __builtin_amdgcn_wmma_f32_16x16x4_f32(
                        false, a, false, b, (short)0, c, false, false);
            }
        }
    }

    // ---- epilogue: transpose D tile through LDS -> coalesced b128 stores.
    //      D layout: VGPR v holds M = v (lo lanes) / v+8 (hi lanes), N = lane%16. ----
    if (oc < OUT_C) {
        const int mbase = hihalf ? 8 : 0;
        #pragma unroll
        for (int v = 0; v < 8; ++v)
            slds[wv][oc][mbase + v] = c[v];
    }
    __syncthreads();

    // lanes 0-3 cover one contiguous 64B line of an output row
    const int soc = lane >> 2;          // 0..7  output channel
    const int seg = lane & 3;           // 0..3  16B segment within the row
    const float4 r = *(const float4*)&slds[wv][soc][seg * 4];
    float4* outp = (float4*)(out +
        ((((long)n * OUT_C + soc) * DHW + z) * DHW + (y0 + wv)) * DHW + x0 + seg * 4);
    *outp = r;
}

extern "C" void kernel_launch(void* const* d_in, const int* in_sizes, int n_in,
                              void* d_out, int out_size, void* d_ws, size_t ws_size,
                              hipStream_t stream) {
    const float* x    = (const float*)d_in[0];
    const float* w    = (const float*)d_in[1];
    const float* bias = (const float*)d_in[2];
    float* out        = (float*)d_out;

    dim3 grid((DHW / TILE_X) * (DHW / TILE_Y), DHW, 2); // (200, 160, 2)
    dim3 block(256);
    conv3d_wmma_f32_kernel<<<grid, block, 0, stream>>>(x, w, bias, out);
}